// Capsule_9706626089109
// MI455X (gfx1250) — compile-verified
//
#include <hip/hip_runtime.h>
#include <hip/hip_bf16.h>

// ---------------------------------------------------------------------------
// Capsule routing on MI455X (gfx1250, wave32):
//   K1  gemm_uhat      : u_hat = u @ W  via V_WMMA_F32_16X16X4_F32
//                        (128x128 block tile, 8 waves, double-buffered LDS,
//                         bank-conflict-free padded tiles, B stored k-minor)
//   K2  vj_kernel      : vj = einsum(softmax(b), u_hat)  (iter0: uniform 1/32)
//   K3  norm_kernel    : rnorm = 1/sqrt(max(sum vj^2, 1e-12))  (global norm)
//   K4  bupd_kernel    : b = einsum(vj*rnorm, u_hat)
//   K5  stats_kernel   : per-(b,n) softmax max + sum-exp over 32 capsules
//   K6  squash_kernel  : out = sqrt(s)/(0.5+s) * vj,  s = sum_d vj^2 + 1e-7
//
// Workspace layout (floats), total ~68.3M floats (~273 MB):
//   uhat [32768][2048], bb [32][32][1024], m [32][1024], s [32][1024],
//   vj [32][32][64], rnorm [1]
// ---------------------------------------------------------------------------

typedef float v2f __attribute__((ext_vector_type(2)));
typedef float v8f __attribute__((ext_vector_type(8)));

#define B_SZ   32
#define N_INP  1024
#define D_INP  512
#define N_CAP  32
#define D_CAPS 64
#define N_OUT  2048                 // N_CAP * D_CAPS
#define M_TOT  (B_SZ * N_INP)       // 32768

// ------------------------------- GEMM --------------------------------------
// C[M_TOT x N_OUT] = A[M_TOT x D_INP] * W[D_INP x N_OUT], fp32 WMMA 16x16x4.
__global__ __launch_bounds__(256) void gemm_uhat(const float* __restrict__ A,
                                                 const float* __restrict__ W,
                                                 float* __restrict__ C) {
    constexpr int LDK = 20;   // padded leading dim: conflict-free, keeps 8B align
    __shared__ __align__(16) float As[2][128][LDK];   // As[buf][m][k]
    __shared__ __align__(16) float Bt[2][128][LDK];   // Bt[buf][n][k]  (k-minor!)

    const int t     = threadIdx.x;
    const int lane  = t & 31;
    const int wid   = t >> 5;
    const int waveM = wid & 3;       // 0..3  -> rows of 32
    const int waveN = wid >> 2;      // 0..1  -> cols of 64
    const int mr    = lane & 15;     // row/col within 16x16 fragment
    const int h     = lane >> 4;     // lane-half selector

    const int mBase = blockIdx.y * 128;
    const int nBase = blockIdx.x * 128;

    // staging-load indices
    const int arow = t >> 2;            // 0..63
    const int acol = (t & 3) * 4;       // 0,4,8,12
    const int brow = t >> 5;            // 0..7   (k within slab)
    const int bcol = (t & 31) * 4;      // 0..124 (n within tile)

    const float* pA0 = A + (size_t)(mBase + arow) * D_INP + acol;
    const float* pA1 = A + (size_t)(mBase + arow + 64) * D_INP + acol;
    const float* pB0 = W + (size_t)brow * N_OUT + nBase + bcol;
    const float* pB1 = W + (size_t)(brow + 8) * N_OUT + nBase + bcol;

    v8f acc[2][4];
#pragma unroll
    for (int i = 0; i < 2; ++i)
#pragma unroll
        for (int j = 0; j < 4; ++j)
#pragma unroll
            for (int e = 0; e < 8; ++e) acc[i][j][e] = 0.0f;

    // ---- prologue: stage slab 0 ----
    float4 a0 = *(const float4*)(pA0);
    float4 a1 = *(const float4*)(pA1);
    float4 b0 = *(const float4*)(pB0);
    float4 b1 = *(const float4*)(pB1);
    *(float4*)&As[0][arow][acol]      = a0;
    *(float4*)&As[0][arow + 64][acol] = a1;
    Bt[0][bcol + 0][brow]     = b0.x;  Bt[0][bcol + 1][brow]     = b0.y;
    Bt[0][bcol + 2][brow]     = b0.z;  Bt[0][bcol + 3][brow]     = b0.w;
    Bt[0][bcol + 0][brow + 8] = b1.x;  Bt[0][bcol + 1][brow + 8] = b1.y;
    Bt[0][bcol + 2][brow + 8] = b1.z;  Bt[0][bcol + 3][brow + 8] = b1.w;
    __syncthreads();

    int buf = 0;
    for (int kt = 0; kt < D_INP; kt += 16) {
        const bool more = (kt + 16 < D_INP);
        if (more) {   // issue next slab's global loads before the WMMA block
            a0 = *(const float4*)(pA0 + kt + 16);
            a1 = *(const float4*)(pA1 + kt + 16);
            b0 = *(const float4*)(pB0 + (size_t)(kt + 16) * N_OUT);
            b1 = *(const float4*)(pB1 + (size_t)(kt + 16) * N_OUT);
        }
        if (kt + 32 < D_INP) {   // warm L2/L0 one slab further ahead
            __builtin_prefetch(pA0 + kt + 32);
            __builtin_prefetch(pB0 + (size_t)(kt + 32) * N_OUT);
        }

#pragma unroll
        for (int ks = 0; ks < 4; ++ks) {
            const int k0 = ks * 4 + 2 * h;   // lane-half picks K {0,1}/{2,3}
            v2f afrag[2], bfrag[4];
#pragma unroll
            for (int mt = 0; mt < 2; ++mt)
                afrag[mt] = *(const v2f*)&As[buf][waveM * 32 + mt * 16 + mr][k0];
#pragma unroll
            for (int nt = 0; nt < 4; ++nt)
                bfrag[nt] = *(const v2f*)&Bt[buf][waveN * 64 + nt * 16 + mr][k0];
#pragma unroll
            for (int nt = 0; nt < 4; ++nt)
#pragma unroll
                for (int mt = 0; mt < 2; ++mt)
                    acc[mt][nt] = __builtin_amdgcn_wmma_f32_16x16x4_f32(
                        false, afrag[mt], false, bfrag[nt], (short)0,
                        acc[mt][nt], false, false);
        }

        if (more) {   // commit next slab into the alternate buffer
            const int nb = buf ^ 1;
            *(float4*)&As[nb][arow][acol]      = a0;
            *(float4*)&As[nb][arow + 64][acol] = a1;
            Bt[nb][bcol + 0][brow]     = b0.x;  Bt[nb][bcol + 1][brow]     = b0.y;
            Bt[nb][bcol + 2][brow]     = b0.z;  Bt[nb][bcol + 3][brow]     = b0.w;
            Bt[nb][bcol + 0][brow + 8] = b1.x;  Bt[nb][bcol + 1][brow + 8] = b1.y;
            Bt[nb][bcol + 2][brow + 8] = b1.z;  Bt[nb][bcol + 3][brow + 8] = b1.w;
        }
        __syncthreads();
        buf ^= 1;
    }

    // C/D layout: VGPR g -> row g + 8*h, col = mr
#pragma unroll
    for (int mt = 0; mt < 2; ++mt)
#pragma unroll
        for (int nt = 0; nt < 4; ++nt) {
            const int row0 = mBase + waveM * 32 + mt * 16 + 8 * h;
            const int col  = nBase + waveN * 64 + nt * 16 + mr;
#pragma unroll
            for (int g = 0; g < 8; ++g)
                C[(size_t)(row0 + g) * N_OUT + col] = acc[mt][nt][g];
        }
}

// ------------------------ vj = einsum(Wv, u_hat) ---------------------------
// One block per (b, o). Threads: d2 = t&31 handles (2*d2, 2*d2+1), 8 n-groups.
__global__ __launch_bounds__(256) void vj_kernel(const float* __restrict__ uhat,
                                                 const float* __restrict__ bb,
                                                 const float* __restrict__ mx,
                                                 const float* __restrict__ se,
                                                 float* __restrict__ vj,
                                                 int useUniform) {
    const int bo = blockIdx.x;          // 0..1023
    const int b  = bo >> 5;
    const int o  = bo & 31;
    const int t  = threadIdx.x;
    const int d2 = t & 31;
    const int ng = t >> 5;              // 0..7

    const float* up = uhat + (size_t)(b * N_INP) * N_OUT + o * D_CAPS + 2 * d2;
    float ax = 0.0f, ay = 0.0f;
    for (int n = ng; n < N_INP; n += 8) {
        float w;
        if (useUniform) {
            w = 1.0f / 32.0f;
        } else {
            const float bv = bb[(b * N_CAP + o) * N_INP + n];
            w = __expf(bv - mx[b * N_INP + n]) / se[b * N_INP + n];
        }
        const float2 u2 = *(const float2*)(up + (size_t)n * N_OUT);
        ax += w * u2.x;
        ay += w * u2.y;
    }
    __shared__ float red[8][64];
    red[ng][2 * d2]     = ax;
    red[ng][2 * d2 + 1] = ay;
    __syncthreads();
    if (t < 64) {
        float sum = 0.0f;
#pragma unroll
        for (int g = 0; g < 8; ++g) sum += red[g][t];
        vj[bo * 64 + t] = sum;
    }
}

// ----------------- global L2 norm of vj (axis=None, incl. batch) -----------
__global__ __launch_bounds__(256) void norm_kernel(const float* __restrict__ vj,
                                                   float* __restrict__ rnorm) {
    const int t = threadIdx.x;
    float s = 0.0f;
    for (int i = t; i < B_SZ * N_CAP * D_CAPS; i += 256) {
        const float v = vj[i];
        s += v * v;
    }
    __shared__ float red[256];
    red[t] = s;
    __syncthreads();
    for (int st = 128; st > 0; st >>= 1) {
        if (t < st) red[t] += red[t + st];
        __syncthreads();
    }
    if (t == 0) rnorm[0] = 1.0f / sqrtf(fmaxf(red[0], 1e-12f));
}

// ------------- b[b,o,n] = sum_d (vj[b,o,d]*rnorm) * u_hat[b,o,n,d] ---------
// One block per (b, o). 8 waves, each wave reduces one n-row (64 floats).
__global__ __launch_bounds__(256) void bupd_kernel(const float* __restrict__ uhat,
                                                   const float* __restrict__ vj,
                                                   const float* __restrict__ rnorm,
                                                   float* __restrict__ bb) {
    const int bo   = blockIdx.x;
    const int b    = bo >> 5;
    const int o    = bo & 31;
    const int t    = threadIdx.x;
    const int lane = t & 31;
    const int wid  = t >> 5;

    const float rn = rnorm[0];
    float2 v2 = *(const float2*)(vj + bo * 64 + 2 * lane);
    v2.x *= rn;
    v2.y *= rn;

    const float* up = uhat + (size_t)(b * N_INP) * N_OUT + o * D_CAPS + 2 * lane;
    for (int n = wid; n < N_INP; n += 8) {
        const float2 u2 = *(const float2*)(up + (size_t)n * N_OUT);
        float p = u2.x * v2.x + u2.y * v2.y;
#pragma unroll
        for (int off = 16; off > 0; off >>= 1) p += __shfl_down(p, off, 32);
        if (lane == 0) bb[(b * N_CAP + o) * N_INP + n] = p;
    }
}

// ---------- per-(b,n) softmax stats over the 32 output capsules ------------
__global__ __launch_bounds__(256) void stats_kernel(const float* __restrict__ bb,
                                                    float* __restrict__ mx,
                                                    float* __restrict__ se) {
    const int idx = blockIdx.x * 256 + threadIdx.x;   // 0..32767
    const int b = idx >> 10;
    const int n = idx & 1023;
    float v[N_CAP];
    float m = -3.402823466e38f;
#pragma unroll
    for (int o = 0; o < N_CAP; ++o) {
        v[o] = bb[(b * N_CAP + o) * N_INP + n];
        m = fmaxf(m, v[o]);
    }
    float s = 0.0f;
#pragma unroll
    for (int o = 0; o < N_CAP; ++o) s += __expf(v[o] - m);
    mx[idx] = m;
    se[idx] = s;
}

// ----------------------------- squash(vj) ----------------------------------
__global__ __launch_bounds__(64) void squash_kernel(const float* __restrict__ vj,
                                                    float* __restrict__ out) {
    const int bo = blockIdx.x;
    const int t  = threadIdx.x;
    __shared__ float red[64];
    const float v = vj[bo * 64 + t];
    red[t] = v * v;
    __syncthreads();
    for (int st = 32; st > 0; st >>= 1) {
        if (t < st) red[t] += red[t + st];
        __syncthreads();
    }
    const float s = red[0] + 1e-7f;
    const float scale = sqrtf(s) / (0.5f + s);
    out[bo * 64 + t] = scale * v;
}

// ---------------------------------------------------------------------------
extern "C" void kernel_launch(void* const* d_in, const int* in_sizes, int n_in,
                              void* d_out, int out_size, void* d_ws, size_t ws_size,
                              hipStream_t stream) {
    const float* u = (const float*)d_in[0];   // [32, 1024, 512]
    const float* W = (const float*)d_in[1];   // [512, 2048]
    float* out = (float*)d_out;               // [32, 32, 64]

    float* ws    = (float*)d_ws;
    float* uhat  = ws;                                   // 67,108,864
    float* bb    = uhat + (size_t)M_TOT * N_OUT;         //  1,048,576
    float* mx    = bb + (size_t)B_SZ * N_CAP * N_INP;    //     32,768
    float* se    = mx + (size_t)B_SZ * N_INP;            //     32,768
    float* vj    = se + (size_t)B_SZ * N_INP;            //     65,536
    float* rnorm = vj + (size_t)B_SZ * N_CAP * D_CAPS;   //          1

    const dim3 gGemm(N_OUT / 128, M_TOT / 128, 1);       // (16, 256)

    // u_hat = u @ W
    gemm_uhat<<<gGemm, 256, 0, stream>>>(u, W, uhat);

    // ---- routing iteration 0: softmax(0) == uniform 1/32 ----
    vj_kernel<<<B_SZ * N_CAP, 256, 0, stream>>>(uhat, bb, mx, se, vj, 1);
    norm_kernel<<<1, 256, 0, stream>>>(vj, rnorm);
    bupd_kernel<<<B_SZ * N_CAP, 256, 0, stream>>>(uhat, vj, rnorm, bb);

    // ---- routing iteration 1 ----
    stats_kernel<<<(B_SZ * N_INP) / 256, 256, 0, stream>>>(bb, mx, se);
    vj_kernel<<<B_SZ * N_CAP, 256, 0, stream>>>(uhat, bb, mx, se, vj, 0);
    norm_kernel<<<1, 256, 0, stream>>>(vj, rnorm);
    bupd_kernel<<<B_SZ * N_CAP, 256, 0, stream>>>(uhat, vj, rnorm, bb);

    // ---- routing iteration 2 ----
    stats_kernel<<<(B_SZ * N_INP) / 256, 256, 0, stream>>>(bb, mx, se);
    vj_kernel<<<B_SZ * N_CAP, 256, 0, stream>>>(uhat, bb, mx, se, vj, 0);

    // ---- output = squash(vj) ----
    squash_kernel<<<B_SZ * N_CAP, 64, 0, stream>>>(vj, out);
}